// PartitionedEncoder_75814762709164
// MI455X (gfx1250) — compile-verified
//
#include <hip/hip_runtime.h>

typedef __attribute__((ext_vector_type(2))) float v2f;
typedef __attribute__((ext_vector_type(8))) float v8f;

#define N_NODES 100000
#define N_EDGES 1600000

__device__ __forceinline__ v8f wmma_f32_16x16x4(v2f a, v2f b, v8f c) {
  // D = A(16x4,f32) * B(4x16,f32) + C(16x16,f32), full-precision matrix path
  return __builtin_amdgcn_wmma_f32_16x16x4_f32(false, a, false, b, (short)0, c,
                                               false, false);
}

__global__ void k_fill(float* __restrict__ p, int n) {
  int i = blockIdx.x * blockDim.x + threadIdx.x;
  int stride = gridDim.x * blockDim.x;
  for (; i < n; i += stride) p[i] = 0.0f;
}

__global__ void k_deg(const long long* __restrict__ ei, float* __restrict__ deg) {
  int e = blockIdx.x * blockDim.x + threadIdx.x;
  if (e < N_EDGES) atomicAdd(&deg[(int)ei[N_EDGES + e]], 1.0f);
}

__global__ void k_dinv(const float* __restrict__ deg, float* __restrict__ dinv) {
  int i = blockIdx.x * blockDim.x + threadIdx.x;
  if (i < N_NODES) dinv[i] = rsqrtf(deg[i] + 1.0f);  // +1 = self loop
}

// Scatter 64 features per edge: acc[dst] += src_feat[src] * dinv[src]*dinv[dst].
// 16 threads per edge, float4 gathers, fp32 global atomics.
__global__ void k_scatter64(const long long* __restrict__ ei,
                            const float* __restrict__ dinv,
                            const float* __restrict__ feat,
                            float* __restrict__ acc) {
  int gid = blockIdx.x * blockDim.x + threadIdx.x;
  int e = gid >> 4;
  int c = gid & 15;
  if (e >= N_EDGES) return;
  int s = (int)ei[e];
  int d = (int)ei[N_EDGES + e];
  float w = dinv[s] * dinv[d];
  const float4 v = ((const float4*)feat)[s * 16 + c];
  float* p = acc + (size_t)d * 64 + c * 4;
  atomicAdd(p + 0, v.x * w);
  atomicAdd(p + 1, v.y * w);
  atomicAdd(p + 2, v.z * w);
  atomicAdd(p + 3, v.w * w);
}

// Fused per-16-node-tile pipeline, one wave per tile, 2 waves per block:
//   A = Xa + X*dinv^2 (self loop)            [16x64]
//   h = elu([A[:, :32]@Wn + bn, A[:,32:]@Ws + bs])   [16x256] -> LDS
//   t = h @ Wf                                [16x64]  -> global
// All matmuls via V_WMMA_F32_16X16X4_F32.
__global__ __launch_bounds__(64) void k_gemm(
    const float* __restrict__ Xa, const float* __restrict__ X,
    const float* __restrict__ dinv,
    const float* __restrict__ Wn, const float* __restrict__ bn,
    const float* __restrict__ Ws, const float* __restrict__ bs,
    const float* __restrict__ Wf, float* __restrict__ t) {
  __shared__ float lds[2][16][260];  // 260-stride padding vs 64 banks

  const int wave = threadIdx.x >> 5;
  const int lane = threadIdx.x & 31;
  const int half = lane >> 4;   // 0: K/M cols 0-1 of quad, 1: cols 2-3
  const int lm   = lane & 15;
  const int tile = blockIdx.x * 2 + wave;   // 6250 tiles exactly
  const int row0 = tile * 16;

  // A fragments for stage 1 (16 k-steps of 4 over the 64 input features),
  // with the self-loop contribution folded in.
  const int arow = row0 + lm;
  const float dv = dinv[arow];
  const float sw = dv * dv;
  const float* xa = Xa + (size_t)arow * 64;
  const float* xr = X + (size_t)arow * 64;
  v2f afrag[16];
#pragma unroll
  for (int kk = 0; kk < 16; ++kk) {
    int c = kk * 4 + half * 2;
    afrag[kk].x = xa[c] + xr[c] * sw;
    afrag[kk].y = xa[c + 1] + xr[c + 1] * sw;
  }

  // Stage 1: branch GEMMs + bias + ELU, h staged to LDS.
#pragma unroll
  for (int br = 0; br < 2; ++br) {
    const float* W = br ? Ws : Wn;
    const float* B = br ? bs : bn;
#pragma unroll
    for (int j = 0; j < 8; ++j) {  // 8 N-tiles of 16 -> 128 cols per branch
      v8f acc = {};
#pragma unroll
      for (int kk = 0; kk < 8; ++kk) {  // K = 32
        int kb = kk * 4 + half * 2;
        v2f bf;
        bf.x = W[(size_t)kb * 128 + j * 16 + lm];
        bf.y = W[(size_t)(kb + 1) * 128 + j * 16 + lm];
        acc = wmma_f32_16x16x4(afrag[br * 8 + kk], bf, acc);
      }
      float bias = B[j * 16 + lm];
#pragma unroll
      for (int r = 0; r < 8; ++r) {
        float x = acc[r] + bias;
        lds[wave][r + half * 8][br * 128 + j * 16 + lm] =
            x > 0.0f ? x : (__expf(x) - 1.0f);  // ELU
      }
    }
  }
  __syncthreads();

  // Stage 2: t = h @ Wf, K = 256 (64 k-steps), 4 output N-tiles.
  v8f acc[4] = {{}, {}, {}, {}};
#pragma unroll 4
  for (int k = 0; k < 64; ++k) {
    int kb = k * 4 + half * 2;
    v2f af;
    af.x = lds[wave][lm][kb];
    af.y = lds[wave][lm][kb + 1];
#pragma unroll
    for (int j = 0; j < 4; ++j) {
      v2f bf;
      bf.x = Wf[(size_t)kb * 64 + j * 16 + lm];
      bf.y = Wf[(size_t)(kb + 1) * 64 + j * 16 + lm];
      acc[j] = wmma_f32_16x16x4(af, bf, acc[j]);
    }
  }
#pragma unroll
  for (int j = 0; j < 4; ++j)
#pragma unroll
    for (int r = 0; r < 8; ++r)
      t[(size_t)(row0 + r + half * 8) * 64 + j * 16 + lm] = acc[j][r];
}

// Per-node softmax over 64 classes; one wave per node, 2 values per lane.
// Folds in the head conv's self-loop term and b_fuse.
__global__ void k_softmax(float* __restrict__ out, const float* __restrict__ t,
                          const float* __restrict__ dinv,
                          const float* __restrict__ bfu) {
  int node = blockIdx.x * 4 + (threadIdx.x >> 5);  // 25000 blocks exactly
  int lane = threadIdx.x & 31;
  float dv = dinv[node];
  float sw = dv * dv;
  size_t base = (size_t)node * 64;
  float v0 = out[base + lane] + t[base + lane] * sw + bfu[lane];
  float v1 = out[base + lane + 32] + t[base + lane + 32] * sw + bfu[lane + 32];
  float m = fmaxf(v0, v1);
#pragma unroll
  for (int off = 16; off > 0; off >>= 1) m = fmaxf(m, __shfl_xor(m, off, 32));
  float e0 = __expf(v0 - m);
  float e1 = __expf(v1 - m);
  float s = e0 + e1;
#pragma unroll
  for (int off = 16; off > 0; off >>= 1) s += __shfl_xor(s, off, 32);
  float inv = 1.0f / s;
  out[base + lane] = e0 * inv;
  out[base + lane + 32] = e1 * inv;
}

extern "C" void kernel_launch(void* const* d_in, const int* in_sizes, int n_in,
                              void* d_out, int out_size, void* d_ws,
                              size_t ws_size, hipStream_t stream) {
  const float* X = (const float*)d_in[0];
  const long long* ei = (const long long*)d_in[1];
  // d_in[2] = batch (all zeros) unused: softmax is per-node
  const float* Wn = (const float*)d_in[3];
  const float* bn = (const float*)d_in[4];
  const float* Ws = (const float*)d_in[5];
  const float* bs = (const float*)d_in[6];
  const float* Wf = (const float*)d_in[7];
  const float* bfu = (const float*)d_in[8];
  float* out = (float*)d_out;

  float* ws = (float*)d_ws;
  float* deg = ws;                                   // N
  float* dinv = ws + N_NODES;                        // N
  float* Xa = ws + 2 * (size_t)N_NODES;              // 64*N
  float* tb = ws + (2 + 64) * (size_t)N_NODES;       // 64*N

  k_fill<<<1024, 256, 0, stream>>>(deg, N_NODES);
  k_fill<<<4096, 256, 0, stream>>>(Xa, N_NODES * 64);
  k_fill<<<4096, 256, 0, stream>>>(out, N_NODES * 64);
  k_deg<<<(N_EDGES + 255) / 256, 256, 0, stream>>>(ei, deg);
  k_dinv<<<(N_NODES + 255) / 256, 256, 0, stream>>>(deg, dinv);
  // Aggregate raw X (linearity: matmul and segment_sum commute) -> 4x less traffic
  k_scatter64<<<(N_EDGES * 16) / 256, 256, 0, stream>>>(ei, dinv, X, Xa);
  // Fused: self-loop + branch GEMMs + bias + ELU + head projection (WMMA f32)
  k_gemm<<<N_NODES / 32, 64, 0, stream>>>(Xa, X, dinv, Wn, bn, Ws, bs, Wf, tb);
  // Aggregate projected 64-dim t (4x less traffic than aggregating h_fused)
  k_scatter64<<<(N_EDGES * 16) / 256, 256, 0, stream>>>(ei, dinv, tb, out);
  k_softmax<<<N_NODES / 4, 128, 0, stream>>>(out, tb, dinv, bfu);
}